// ScaledDotProductWithEdgeAttention_33543694582411
// MI455X (gfx1250) — compile-verified
//
#include <hip/hip_runtime.h>

#define NB  4
#define NH  8
#define NN  256
#define DQ  80      // NK_DIM + EK_DIM
#define NKD 64
#define EKD 16
#define DVD 64
#define NNP (NN + 4)   // padded LDS row stride (260 % 64 = 4 -> conflict-free)

typedef __attribute__((ext_vector_type(2))) float v2f;
typedef __attribute__((ext_vector_type(8))) float v8f;

#if defined(__has_builtin)
#  if __has_builtin(__builtin_amdgcn_global_load_async_to_lds_b32) && \
      __has_builtin(__builtin_amdgcn_s_wait_asynccnt)
#    define USE_ASYNC_LDS 1
#  endif
#endif
#ifndef USE_ASYNC_LDS
#  define USE_ASYNC_LDS 0
#endif

#if USE_ASYNC_LDS
typedef __attribute__((address_space(1))) int as1_int;   // global
typedef __attribute__((address_space(3))) int as3_int;   // LDS
#endif

__launch_bounds__(32)
__global__ void gta_kernel(const float* __restrict__ q,
                           const float* __restrict__ nk,
                           const float* __restrict__ ek,
                           const float* __restrict__ v,
                           const int*   __restrict__ mask,
                           float* __restrict__ out,
                           float* __restrict__ attn_out)
{
    __shared__ float sS[16 * NNP];     // ~16.6KB: score/prob tile (16 q x 256 k, padded)
    __shared__ float sQE[16 * EKD];    // 1KB: edge-query tile

    const int lane = threadIdx.x;            // 0..31 (wave32)
    const int bh   = blockIdx.x >> 4;        // 0..31 == b*H + h
    const int qt   = blockIdx.x & 15;
    const int q0   = qt * 16;
    const int b    = bh >> 3;

    const size_t qrow0 = (size_t)bh * NN + q0;   // first query row, head-flat
    const size_t krow0 = (size_t)bh * NN;        // first key row, head-flat

    // ---- stage qe tile (16x16) into LDS: async-copy path when available ----
#if USE_ASYNC_LDS
    for (int i = lane; i < 16 * EKD; i += 32) {
        int m = i >> 4, e = i & 15;
        void* gsrc = const_cast<float*>(q + (qrow0 + m) * DQ + NKD + e);
        __builtin_amdgcn_global_load_async_to_lds_b32(
            (as1_int*)gsrc, (as3_int*)(sQE + i), 0, 0);
    }
#else
    for (int i = lane; i < 16 * EKD; i += 32) {
        int m = i >> 4, e = i & 15;
        sQE[i] = q[(qrow0 + m) * DQ + NKD + e];
    }
#endif

    // ---- preload qn A-fragments: 16 K-chunks of depth 4 (ISA 16x4 A layout) ----
    const int koff = (lane < 16) ? 0 : 2;
    const int mn   = lane & 15;
    v2f aq[16];
#pragma unroll
    for (int kk = 0; kk < 16; ++kk)
        aq[kk] = *(const v2f*)(q + (qrow0 + mn) * DQ + kk * 4 + koff);

#if USE_ASYNC_LDS
    __builtin_amdgcn_s_wait_asynccnt(0);
#endif
    __syncthreads();  // 1-wave WG: lowers to S_NOP, documents the LDS hand-off

    // ---- scores: node via WMMA f32 (batched B-frag loads), edge via 16-dot ----
    for (int kt = 0; kt < 16; ++kt) {
        const int k0 = kt * 16;
        const float* nkrow = nk + (krow0 + k0 + mn) * NKD + koff;
        v8f c = {};
#pragma unroll
        for (int kg = 0; kg < 2; ++kg) {          // two batches of 8 -> 8 loads in flight
            v2f bb[8];
#pragma unroll
            for (int j = 0; j < 8; ++j)
                bb[j] = *(const v2f*)(nkrow + (kg * 8 + j) * 4);
#pragma unroll
            for (int j = 0; j < 8; ++j)
                c = __builtin_amdgcn_wmma_f32_16x16x4_f32(false, aq[kg * 8 + j], false,
                                                          bb[j], (short)0, c, false, false);
        }
#pragma unroll
        for (int r = 0; r < 8; ++r) {
            const int m = r + ((lane < 16) ? 0 : 8);   // C/D layout row
            const int n = mn;                          // C/D layout col
            // edge term: dot(qe[m,:16], ek[m, k0+n, :16]) -- 64B contiguous per lane
            const float4* ep = (const float4*)(ek +
                (((size_t)bh * NN + q0 + m) * NN + (k0 + n)) * EKD);
            // prefetch this lane's line for the NEXT k-tile (+1KB) while WMMA/FMAs run
            if (kt < 15) __builtin_prefetch(ep + 64, 0, 3);
            const float4* qp = (const float4*)(sQE + m * EKD);
            float acc = 0.f;
#pragma unroll
            for (int e4 = 0; e4 < 4; ++e4) {
                float4 ev = ep[e4], qv = qp[e4];
                acc += ev.x * qv.x + ev.y * qv.y + ev.z * qv.z + ev.w * qv.w;
            }
            float s = (c[r] + acc) * 0.125f;           // / TEMPERATURE
            if (mask[((size_t)b * NN + q0 + m) * NN + (k0 + n)] == 0)
                s = -1.0e9f;                            // NEG_INF after scaling
            sS[m * NNP + k0 + n] = s;
        }
    }
    __syncthreads();

    // ---- softmax: each lane owns half of one row (float4 LDS traffic) ----
    {
        const int row = lane & 15;
        float4* rp = (float4*)(sS + row * NNP) + (lane >> 4) * 32;  // 32 float4 = 128 floats
        float mx = -3.4e38f;
        for (int c1 = 0; c1 < 32; ++c1) {
            float4 sv = rp[c1];
            mx = fmaxf(mx, fmaxf(fmaxf(sv.x, sv.y), fmaxf(sv.z, sv.w)));
        }
        mx = fmaxf(mx, __shfl_xor(mx, 16, 32));
        float sum = 0.f;
        for (int c1 = 0; c1 < 32; ++c1) {
            float4 sv = rp[c1];
            sv.x = __expf(sv.x - mx);
            sv.y = __expf(sv.y - mx);
            sv.z = __expf(sv.z - mx);
            sv.w = __expf(sv.w - mx);
            rp[c1] = sv;
            sum += sv.x + sv.y + sv.z + sv.w;
        }
        sum += __shfl_xor(sum, 16, 32);
        const float inv = 1.0f / sum;
        for (int c1 = 0; c1 < 32; ++c1) {
            float4 sv = rp[c1];
            sv.x *= inv; sv.y *= inv; sv.z *= inv; sv.w *= inv;
            rp[c1] = sv;
        }
    }
    __syncthreads();

    // ---- write attn tile (required output), coalesced float4 per row ----
    for (int i = lane; i < 16 * (NN / 4); i += 32) {
        const int rr = i >> 6, cc = i & 63;            // 64 float4 per row
        ((float4*)(attn_out + (qrow0 + rr) * NN))[cc] =
            ((const float4*)(sS + rr * NNP))[cc];
    }

    // ---- O(16x64) = P(16x256) x V(256x64) via WMMA f32, batched frag loads ----
#pragma unroll
    for (int nt = 0; nt < 4; ++nt) {
        v8f c = {};
        for (int kg = 0; kg < 8; ++kg) {               // groups of 8 K-chunks
            v2f aa[8], bb[8];
#pragma unroll
            for (int j = 0; j < 8; ++j) {
                const int kc = kg * 8 + j;
                aa[j] = *(const v2f*)(sS + mn * NNP + kc * 4 + koff);
                const float* vp = v + (krow0 + kc * 4 + koff) * DVD + nt * 16 + mn;
                bb[j].x = vp[0];
                bb[j].y = vp[DVD];
            }
#pragma unroll
            for (int j = 0; j < 8; ++j)
                c = __builtin_amdgcn_wmma_f32_16x16x4_f32(false, aa[j], false, bb[j],
                                                          (short)0, c, false, false);
        }
#pragma unroll
        for (int r = 0; r < 8; ++r) {
            const int m = r + ((lane < 16) ? 0 : 8);
            out[(qrow0 + m) * DVD + nt * 16 + mn] = c[r];
        }
    }
}

extern "C" void kernel_launch(void* const* d_in, const int* in_sizes, int n_in,
                              void* d_out, int out_size, void* d_ws, size_t ws_size,
                              hipStream_t stream) {
    const float* q    = (const float*)d_in[0];
    const float* nk   = (const float*)d_in[1];
    const float* ek   = (const float*)d_in[2];
    const float* v    = (const float*)d_in[3];
    const int*   mask = (const int*)d_in[4];

    float* out  = (float*)d_out;                              // (B,H,N,DV)
    float* attn = out + (size_t)NB * NH * NN * DVD;           // (B,H,N,N)

    dim3 grid(NB * NH * (NN / 16));   // 512 single-wave workgroups
    gta_kernel<<<grid, 32, 0, stream>>>(q, nk, ek, v, mask, out, attn);
}